// ptBEVnet_80874234183864
// MI455X (gfx1250) — compile-verified
//
#include <hip/hip_runtime.h>
#include <hip/hip_bf16.h>
#include <math.h>

// ---------------------------------------------------------------------------
// ptBEVnet on MI455X (gfx1250): per-point MLP via V_WMMA_F32_16X16X4_F32,
// fused BN-stat accumulation, atomic segment-max pooling, 3x3 BEV max-pool.
// Round 2: fold addresses to base+immediate-offset form, hoist bias/W bases.
// ---------------------------------------------------------------------------

typedef __attribute__((ext_vector_type(2))) float v2f;
typedef __attribute__((ext_vector_type(8))) float v8f;

#define NPTS   240000
#define FEA    9
#define RES    2
#define INSTR  (FEA + RES)          // 11 floats per point row
#define BB     2
#define GX     480
#define GY     360
#define NSEG   (BB * GX * GY)       // 345600
#define NTILES (NPTS / 16)          // 15000 (exact)
#define EPSBN  1e-5f

// ---------------- float atomic max via int/uint trick ----------------------
__device__ __forceinline__ void atomicMaxF(float* addr, float val) {
    if (!(val == val)) return;                 // skip NaN
    if (val >= 0.0f) atomicMax((int*)addr, __float_as_int(val));
    else             atomicMin((unsigned int*)addr, __float_as_uint(val));
}

// ---------------- fill kernel (e.g. -inf init) ------------------------------
__global__ void fill_f32_kernel(float* __restrict__ p, size_t n, float v) {
    size_t i = (size_t)blockIdx.x * blockDim.x + threadIdx.x;
    size_t stride = (size_t)gridDim.x * blockDim.x;
    for (; i < n; i += stride) p[i] = v;
}

// ---------------- bn0 statistics over first 9 feature columns --------------
__global__ void bn0_stats_kernel(const float* __restrict__ fea,
                                 float* __restrict__ raw /* sum@[c], sq@[16+c] */) {
    __shared__ float s[32];
    const int tid = threadIdx.x;
    if (tid < 32) s[tid] = 0.0f;
    __syncthreads();

    float ls[FEA], lq[FEA];
#pragma unroll
    for (int c = 0; c < FEA; ++c) { ls[c] = 0.0f; lq[c] = 0.0f; }

    size_t stride = (size_t)gridDim.x * blockDim.x;
    for (size_t pt = (size_t)blockIdx.x * blockDim.x + tid; pt < NPTS; pt += stride) {
        const float* row = fea + pt * INSTR;
#pragma unroll
        for (int c = 0; c < FEA; ++c) {
            float v = row[c];
            ls[c] += v; lq[c] += v * v;
        }
    }
#pragma unroll
    for (int c = 0; c < FEA; ++c) {
        atomicAdd(&s[c], ls[c]);
        atomicAdd(&s[16 + c], lq[c]);
    }
    __syncthreads();
    if (tid < 32) atomicAdd(&raw[tid], s[tid]);
}

// ---------------- finalize BN stats: (sum, sumsq) -> (mean, rstd) ----------
__global__ void finalize_bn_kernel(const float* __restrict__ raw,
                                   float* __restrict__ fin, int d, int pad) {
    int t = threadIdx.x;
    if (t < d) {
        const float invN = 1.0f / (float)NPTS;
        float mean = raw[t] * invN;
        float var  = raw[pad + t] * invN - mean * mean;
        fin[t]       = mean;
        fin[pad + t] = rsqrtf(var + EPSBN);
    }
}

// ---------------- voxel ids, counts, residual segment-max ------------------
__global__ void points_kernel(const float* __restrict__ fea,
                              const int* __restrict__ ind,
                              int* __restrict__ vox,
                              int* __restrict__ cnt,
                              float* __restrict__ pres) {
    size_t stride = (size_t)gridDim.x * blockDim.x;
    for (size_t pt = (size_t)blockIdx.x * blockDim.x + threadIdx.x; pt < NPTS; pt += stride) {
        int b = ind[pt * 3 + 0];
        int x = ind[pt * 3 + 1];
        int y = ind[pt * 3 + 2];
        int v = (b * GX + x) * GY + y;
        vox[pt] = v;
        atomicAdd(&cnt[v], 1);
        atomicMaxF(&pres[(size_t)v * RES + 0], fea[pt * INSTR + FEA + 0]);
        atomicMaxF(&pres[(size_t)v * RES + 1], fea[pt * INSTR + FEA + 1]);
    }
}

// ---------------- WMMA MLP layer ------------------------------------------
// Each block: 128 threads = 4 waves. Grid-stride over 16-point tiles.
// Stage BN(+ReLU)-transformed input tile in LDS, each wave computes 16x16
// output tiles with chained v_wmma_f32_16x16x4_f32 over K.
template <int DIN, int DOUT, bool RELU_IN, bool LAST>
__launch_bounds__(128)
__global__ void mlp_layer_kernel(const float* __restrict__ inAct, int inStride,
                                 const float* __restrict__ finIn, int statStride,
                                 const float* __restrict__ gIn,
                                 const float* __restrict__ bIn,
                                 const float* __restrict__ W,
                                 const float* __restrict__ bias,
                                 float* __restrict__ outPre,
                                 float* __restrict__ rawOut,
                                 const int* __restrict__ vox,
                                 float* __restrict__ pooled) {
    constexpr int DINP = (DIN + 3) & ~3;   // K padded to multiple of 4
    constexpr int LDSP = DINP + 4;         // LDS row stride (bank-conflict pad)
    constexpr int NCT  = DOUT / 16;        // column tiles
    constexpr int PER  = (NCT + 3) / 4;    // max column tiles per wave

    __shared__ float tile[16 * LDSP];
    __shared__ float sSum[DOUT];
    __shared__ float sSq[DOUT];
    __shared__ int   sVox[16];

    const int tid  = threadIdx.x;
    const int wave = tid >> 5;
    const int lane = tid & 31;
    const int half = lane >> 4;   // 0: lanes 0-15, 1: lanes 16-31
    const int l16  = lane & 15;

    // per-wave column tiles: hoist col index + bias out of the tile loop
    int   colR[PER];
    float biasR[PER];
    int nct = 0;
    for (int n = wave; n < NCT; n += 4) {
        colR[nct]  = n * 16 + l16;
        biasR[nct] = bias[colR[nct]];
        ++nct;
    }

    if (!LAST) {
        for (int i = tid; i < DOUT; i += 128) { sSum[i] = 0.0f; sSq[i] = 0.0f; }
    }
    __syncthreads();

    // loop-invariant A-fragment base in LDS
    const float* tp = &tile[l16 * LDSP + half * 2];

    for (int t = blockIdx.x; t < NTILES; t += gridDim.x) {
        const int ptBase = t * 16;

        // ---- stage input tile: apply previous layer's BN (+ReLU) ----
        for (int i = tid; i < 16 * DINP; i += 128) {
            int r = i / DINP, c = i - r * DINP;
            float v = 0.0f;
            if (c < DIN) {
                float x  = inAct[(size_t)(ptBase + r) * inStride + c];
                float mu = finIn[c];
                float rs = finIn[statStride + c];
                v = (x - mu) * rs * gIn[c] + bIn[c];
                if (RELU_IN) v = fmaxf(v, 0.0f);
            }
            tile[r * LDSP + c] = v;
        }
        if (LAST && tid < 16) sVox[tid] = vox[ptBase + tid];
        __syncthreads();

        // ---- each wave: 16x16 output tiles over its share of columns ----
        for (int j = 0; j < nct; ++j) {
            const int col = colR[j];
            // loop-invariant B-fragment base: immediate offsets k0*DOUT below
            const float* wp = W + (size_t)(half * 2) * DOUT + col;

            v8f acc = {};
#pragma unroll
            for (int k0 = 0; k0 < DINP; k0 += 4) {
                v2f a, b;
                a.x = tp[k0];
                a.y = tp[k0 + 1];
                if (DINP == DIN) {
                    b.x = wp[(size_t)k0 * DOUT];
                    b.y = wp[(size_t)k0 * DOUT + DOUT];
                } else {           // layer 1 only: K padded 9 -> 12
                    const int kk = k0 + half * 2;
                    b.x = (kk < DIN)     ? wp[(size_t)k0 * DOUT]        : 0.0f;
                    b.y = (kk + 1 < DIN) ? wp[(size_t)k0 * DOUT + DOUT] : 0.0f;
                }
                acc = __builtin_amdgcn_wmma_f32_16x16x4_f32(
                    false, a, false, b, (short)0, acc, false, false);
            }

            const float bcol = biasR[j];
            if (LAST) {
#pragma unroll
                for (int r = 0; r < 8; ++r) {
                    const int m = r + half * 8;    // C/D layout: VGPR r -> M=r (+8)
                    atomicMaxF(&pooled[(size_t)sVox[m] * DOUT + col], acc[r] + bcol);
                }
            } else {
                // one base address; stores use immediate offsets r*DOUT*4
                float* op = outPre + (size_t)(ptBase + half * 8) * DOUT + col;
                float ps = 0.0f, pq = 0.0f;
#pragma unroll
                for (int r = 0; r < 8; ++r) {
                    float y = acc[r] + bcol;
                    op[(size_t)r * DOUT] = y;
                    ps += y; pq += y * y;
                }
                atomicAdd(&sSum[col], ps);
                atomicAdd(&sSq[col], pq);
            }
        }
        __syncthreads();
    }

    if (!LAST) {
        for (int i = tid; i < DOUT; i += 128) {
            atomicAdd(&rawOut[i], sSum[i]);
            atomicAdd(&rawOut[DOUT + i], sSq[i]);
        }
    }
}

// ---------------- 3x3 max-pool + occupancy mask + concat -------------------
// One block per (b, x); 256 threads = 256 channels; sweep y.
// Neighbor accesses are center-base + compile-time immediate offsets.
__global__ void out_kernel(const float* __restrict__ pooled,
                           const float* __restrict__ pres,
                           const int* __restrict__ cnt,
                           float* __restrict__ out) {
    const int bb = blockIdx.x / GX;
    const int x  = blockIdx.x - bb * GX;
    const int c  = threadIdx.x;

    const int vcol = (bb * GX + x) * GY;              // center column base
    float* outc = out + (((size_t)bb * 258 + c) * GX + x) * GY;

    for (int y = 0; y < GY; ++y) {
        const float* pc = pooled + ((size_t)(vcol + y)) * 256 + c;  // center
        const int*   cc = cnt + vcol + y;
        float m = -INFINITY;
#pragma unroll
        for (int dx = -1; dx <= 1; ++dx) {
            const int xx = x + dx;
            if (xx < 0 || xx >= GX) continue;         // uniform per block
#pragma unroll
            for (int dy = -1; dy <= 1; ++dy) {
                const int yy = y + dy;
                if (yy < 0 || yy >= GY) continue;
                const int d = dx * GY + dy;           // compile-time constant
                const float val = (cc[d] > 0) ? pc[(long)d * 256] : 0.0f;
                m = fmaxf(m, val);
            }
        }
        outc[y] = m;

        if (c < RES) {
            const float rv = (cc[0] > 0) ? pres[(size_t)(vcol + y) * RES + c] : 0.0f;
            out[(((size_t)bb * 258 + 256 + c) * GX + x) * GY + y] = rv;
        }
    }
}

// ---------------------------------------------------------------------------
extern "C" void kernel_launch(void* const* d_in, const int* in_sizes, int n_in,
                              void* d_out, int out_size, void* d_ws, size_t ws_size,
                              hipStream_t stream) {
    const float* fea  = (const float*)d_in[0];
    const int*   ind  = (const int*)d_in[1];
    const float* bn0g = (const float*)d_in[2];
    const float* bn0b = (const float*)d_in[3];
    const float* w1   = (const float*)d_in[4];
    const float* b1   = (const float*)d_in[5];
    const float* bn1g = (const float*)d_in[6];
    const float* bn1b = (const float*)d_in[7];
    const float* w2   = (const float*)d_in[8];
    const float* b2   = (const float*)d_in[9];
    const float* bn2g = (const float*)d_in[10];
    const float* bn2b = (const float*)d_in[11];
    const float* w3   = (const float*)d_in[12];
    const float* b3   = (const float*)d_in[13];
    const float* bn3g = (const float*)d_in[14];
    const float* bn3b = (const float*)d_in[15];
    const float* w4   = (const float*)d_in[16];
    const float* b4   = (const float*)d_in[17];
    float* out = (float*)d_out;

    // ---- workspace layout (256B aligned slabs) ----
    char* ws = (char*)d_ws;
    size_t off = 0;
    auto take = [&](size_t bytes) -> char* {
        char* p = ws + off;
        off += (bytes + 255) & ~(size_t)255;
        return p;
    };
    float* raw0  = (float*)take(32 * sizeof(float));      // sum@[c], sq@[16+c]
    float* fin0  = (float*)take(32 * sizeof(float));
    float* raw1  = (float*)take(64 * sizeof(float));
    float* fin1  = (float*)take(64 * sizeof(float));
    float* raw2  = (float*)take(128 * sizeof(float));
    float* fin2  = (float*)take(128 * sizeof(float));
    float* raw3  = (float*)take(256 * sizeof(float));
    float* fin3  = (float*)take(256 * sizeof(float));
    int*   vox   = (int*)take((size_t)NPTS * sizeof(int));
    int*   cnt   = (int*)take((size_t)NSEG * sizeof(int));
    float* pres  = (float*)take((size_t)NSEG * RES * sizeof(float));
    float* hpre1 = (float*)take((size_t)NPTS * 32 * sizeof(float));
    float* hpre2 = (float*)take((size_t)NPTS * 64 * sizeof(float));
    float* hpre3 = (float*)take((size_t)NPTS * 128 * sizeof(float));
    float* pooled = (float*)take((size_t)NSEG * 256 * sizeof(float));
    (void)ws_size; (void)in_sizes; (void)n_in; (void)out_size;

    // ---- init ----
    hipMemsetAsync(raw0, 0, 32 * sizeof(float), stream);
    hipMemsetAsync(raw1, 0, 64 * sizeof(float), stream);
    hipMemsetAsync(raw2, 0, 128 * sizeof(float), stream);
    hipMemsetAsync(raw3, 0, 256 * sizeof(float), stream);
    hipMemsetAsync(cnt, 0, (size_t)NSEG * sizeof(int), stream);
    fill_f32_kernel<<<4096, 256, 0, stream>>>(pooled, (size_t)NSEG * 256, -INFINITY);
    fill_f32_kernel<<<512, 256, 0, stream>>>(pres, (size_t)NSEG * RES, -INFINITY);

    // ---- voxel ids / counts / residual pooling ----
    points_kernel<<<938, 256, 0, stream>>>(fea, ind, vox, cnt, pres);

    // ---- bn0 stats ----
    bn0_stats_kernel<<<512, 256, 0, stream>>>(fea, raw0);
    finalize_bn_kernel<<<1, 128, 0, stream>>>(raw0, fin0, FEA, 16);

    // ---- MLP layers (WMMA f32 16x16x4) ----
    mlp_layer_kernel<9, 32, false, false><<<1024, 128, 0, stream>>>(
        fea, INSTR, fin0, 16, bn0g, bn0b, w1, b1, hpre1, raw1, nullptr, nullptr);
    finalize_bn_kernel<<<1, 128, 0, stream>>>(raw1, fin1, 32, 32);

    mlp_layer_kernel<32, 64, true, false><<<1024, 128, 0, stream>>>(
        hpre1, 32, fin1, 32, bn1g, bn1b, w2, b2, hpre2, raw2, nullptr, nullptr);
    finalize_bn_kernel<<<1, 128, 0, stream>>>(raw2, fin2, 64, 64);

    mlp_layer_kernel<64, 128, true, false><<<1024, 128, 0, stream>>>(
        hpre2, 64, fin2, 64, bn2g, bn2b, w3, b3, hpre3, raw3, nullptr, nullptr);
    finalize_bn_kernel<<<1, 128, 0, stream>>>(raw3, fin3, 128, 128);

    mlp_layer_kernel<128, 256, true, true><<<1024, 128, 0, stream>>>(
        hpre3, 128, fin3, 128, bn3g, bn3b, w4, b4, nullptr, nullptr, vox, pooled);

    // ---- 3x3 max-pool + occupancy + concat ----
    out_kernel<<<BB * GX, 256, 0, stream>>>(pooled, pres, cnt, out);
}